// GCN_Body_6296422056681
// MI455X (gfx1250) — compile-verified
//
#include <hip/hip_runtime.h>

// GCN conv: out = D^-1/2 (A+I) D^-1/2 (x W) + b
// N=100000 nodes, E=1600000 edges, F=128 features.
//
// MI455X reasoning: GEMM is bandwidth-trivial (51MB in / 51MB out), so keep
// full fp32 precision using V_WMMA_F32_16X16X4_F32. h' (51.2MB) fits in the
// 192MB L2, so the edge-phase random gathers + fp32 atomics are L2-resident;
// native agent-scope global_atomic_add_f32 is the right scatter path.
//
// W is staged in LDS K-pair-interleaved (float2 {W[k][n], W[k+1][n]}) so each
// B fragment is ONE aligned ds_load_b64 into an even VGPR pair (WMMA sources
// must be even-aligned) -- no v_mov repacking. Pair-row stride 144 float2:
// (288p + 2n) mod 64 puts the two half-waves in disjoint 32-bank blocks.

#define NF 128
#define PSTR 144  // float2 stride per K-pair row of W in LDS

typedef __attribute__((ext_vector_type(2))) float v2f;
typedef __attribute__((ext_vector_type(8))) float v8f;

__global__ void deg_init_kernel(float* __restrict__ deg, int n) {
  int i = blockIdx.x * blockDim.x + threadIdx.x;
  if (i < n) deg[i] = 1.0f;  // self loop contributes 1 to every degree
}

__global__ void deg_count_kernel(const int* __restrict__ ei,
                                 float* __restrict__ deg, int nedges) {
  int e = blockIdx.x * blockDim.x + threadIdx.x;
  if (e < nedges) {
    int dst = ei[nedges + e];  // row 1 of edge_index
    __hip_atomic_fetch_add(&deg[dst], 1.0f, __ATOMIC_RELAXED,
                           __HIP_MEMORY_SCOPE_AGENT);
  }
}

__global__ void rsqrt_kernel(float* __restrict__ deg, int n) {
  int i = blockIdx.x * blockDim.x + threadIdx.x;
  if (i < n) deg[i] = rsqrtf(deg[i]);  // deg >= 1 always (self loop)
}

// h' = (x @ W) * dinv[row]; also written into d_out as the self-loop term.
// One wave per 16-row tile; 8 waves per block; W staged in LDS (pair layout).
__global__ __launch_bounds__(256)
void gemm_scale_kernel(const float* __restrict__ x, const float* __restrict__ W,
                       const float* __restrict__ dinv, float* __restrict__ hp,
                       float* __restrict__ outinit, int nrows) {
  __shared__ v2f Wlds[(NF / 2) * PSTR];  // 64 pair-rows x 144 float2 = 72KB
  for (int i = threadIdx.x; i < NF * NF; i += 256) {
    int k = i >> 7, c = i & 127;
    Wlds[(k >> 1) * PSTR + c][k & 1] = W[i];
  }
  __syncthreads();

  const int wave = threadIdx.x >> 5;
  const int lane = threadIdx.x & 31;
  const int m0 = (blockIdx.x * 8 + wave) * 16;
  if (m0 >= nrows) return;  // wave-uniform: EXEC stays all-ones for WMMA

  const int half  = lane >> 4;   // 0 = lanes 0-15, 1 = lanes 16-31
  const int khalf = half * 2;    // A/B K sub-offset within each 4-wide step
  const int colb  = lane & 15;

  v8f acc[8] = {};

  // A fragment: row m0+colb, cols {kk+khalf, kk+khalf+1} -> one b64 load
  const float* xr = x + (size_t)(m0 + colb) * NF + khalf;

  for (int kk = 0; kk < NF; kk += 4) {
    v2f a = *(const v2f*)(xr + kk);
    // B fragments: pair-row p = (kk+khalf)/2 = kk/2 + half
    const v2f* wr = &Wlds[((kk >> 1) + half) * PSTR + colb];
#pragma unroll
    for (int nt = 0; nt < 8; ++nt) {  // 8 x 16-col output tiles = 128 cols
      v2f b = wr[nt * 16];            // {W[k][n], W[k+1][n]} in one ds_load_b64
      acc[nt] = __builtin_amdgcn_wmma_f32_16x16x4_f32(
          false, a, false, b, (short)0, acc[nt], false, false);
    }
  }

  // C/D layout: VGPR r -> row m0 + r + 8*half, col nt*16 + colb
  const int rbase = half << 3;
#pragma unroll
  for (int r = 0; r < 8; ++r) {
    const int row = m0 + rbase + r;
    const float dv = dinv[row];
    const size_t ro = (size_t)row * NF + colb;
#pragma unroll
    for (int nt = 0; nt < 8; ++nt) {
      float v = acc[nt][r] * dv;
      hp[ro + nt * 16] = v;
      outinit[ro + nt * 16] = v;  // self-loop term seeds the accumulator
    }
  }
}

// One wave per edge: float4 gather of h'[src], 4 fp32 L2 atomics into out[dst].
__global__ void scatter_kernel(const int* __restrict__ ei,
                               const float* __restrict__ hp,
                               float* __restrict__ out, int nedges) {
  int t = blockIdx.x * blockDim.x + threadIdx.x;
  int e = t >> 5;
  if (e >= nedges) return;
  int lane = t & 31;
  int src = ei[e];
  int dst = ei[nedges + e];
  const float4 v = *(const float4*)(hp + (size_t)src * NF + lane * 4);
  float* o = out + (size_t)dst * NF + lane * 4;
  __hip_atomic_fetch_add(o + 0, v.x, __ATOMIC_RELAXED, __HIP_MEMORY_SCOPE_AGENT);
  __hip_atomic_fetch_add(o + 1, v.y, __ATOMIC_RELAXED, __HIP_MEMORY_SCOPE_AGENT);
  __hip_atomic_fetch_add(o + 2, v.z, __ATOMIC_RELAXED, __HIP_MEMORY_SCOPE_AGENT);
  __hip_atomic_fetch_add(o + 3, v.w, __ATOMIC_RELAXED, __HIP_MEMORY_SCOPE_AGENT);
}

__global__ void finalize_kernel(float* __restrict__ out,
                                const float* __restrict__ dinv,
                                const float* __restrict__ bias,
                                long long total) {
  long long t = (long long)blockIdx.x * blockDim.x + threadIdx.x;
  if (t < total) {
    int row = (int)(t >> 7);
    int f = (int)(t & 127);
    out[t] = out[t] * dinv[row] + bias[f];
  }
}

extern "C" void kernel_launch(void* const* d_in, const int* in_sizes, int n_in,
                              void* d_out, int out_size, void* d_ws, size_t ws_size,
                              hipStream_t stream) {
  const float* x  = (const float*)d_in[0];
  const int*   ei = (const int*)d_in[1];   // [2, E] flat: row0=src, row1=dst
  const float* W  = (const float*)d_in[2];
  const float* b  = (const float*)d_in[3];
  float* out = (float*)d_out;

  const int N = in_sizes[0] / NF;
  const int E = in_sizes[1] / 2;

  // Scratch: dinv[N] then 512B-aligned h'[N*128]
  float* deg = (float*)d_ws;
  size_t off = ((size_t)N * sizeof(float) + 511) & ~(size_t)511;
  float* hp = (float*)((char*)d_ws + off);

  deg_init_kernel<<<(N + 255) / 256, 256, 0, stream>>>(deg, N);
  deg_count_kernel<<<(E + 255) / 256, 256, 0, stream>>>(ei, deg, E);
  rsqrt_kernel<<<(N + 255) / 256, 256, 0, stream>>>(deg, N);

  // 8 waves/block x 16 rows/wave = 128 rows per block
  gemm_scale_kernel<<<(N + 127) / 128, 256, 0, stream>>>(x, W, deg, hp, out, N);

  // 32 lanes per edge -> E*32 threads
  long long sthreads = (long long)E * 32;
  scatter_kernel<<<(int)((sthreads + 255) / 256), 256, 0, stream>>>(ei, hp, out, E);

  long long total = (long long)N * NF;
  finalize_kernel<<<(int)((total + 255) / 256), 256, 0, stream>>>(out, deg, b, total);
}